// LTCCell_32074815766640
// MI455X (gfx1250) — compile-verified
//
#include <hip/hip_runtime.h>

#ifndef __has_builtin
#define __has_builtin(x) 0
#endif

#define B_SZ 1024
#define I_SZ 64
#define H_SZ 256
#define UNFOLDS 6
#define TB 32
#define TJ 8
#define NTHREADS 256

typedef float f2 __attribute__((ext_vector_type(2)));
typedef float f4 __attribute__((ext_vector_type(4)));
typedef float f8 __attribute__((ext_vector_type(8)));
typedef int   i4v __attribute__((vector_size(4 * sizeof(int))));  // matches builtin param

#define AS1 __attribute__((address_space(1)))
#define AS3 __attribute__((address_space(3)))

#if __has_builtin(__builtin_amdgcn_wmma_f32_16x16x4_f32)
#define HAVE_WMMA4 1
#else
#define HAVE_WMMA4 0
#endif

__device__ __forceinline__ float fast_tanh(float x) {
#if __has_builtin(__builtin_amdgcn_tanhf)
  return __builtin_amdgcn_tanhf(x);           // v_tanh_f32 (gfx1250 trans op)
#else
  // tanh(x) = 2/(1+exp2(-2x*log2(e))) - 1
  float e = __builtin_amdgcn_exp2f(x * -2.88539008177793f);
  float r = __builtin_amdgcn_rcpf(1.0f + e);
  return __builtin_fmaf(2.0f, r, -1.0f);
#endif
}

__device__ __forceinline__ void async_b128(const void* g, void* l) {
#if __has_builtin(__builtin_amdgcn_global_load_async_to_lds_b128)
  __builtin_amdgcn_global_load_async_to_lds_b128((AS1 i4v*)g, (AS3 i4v*)l, 0, 0);
#else
  unsigned loff = (unsigned)(unsigned long long)(AS3 char*)l;  // LDS byte offset
  asm volatile("global_load_async_to_lds_b128 %0, %1, off" :: "v"(loff), "v"(g) : "memory");
#endif
}

__device__ __forceinline__ void async_b32(const void* g, void* l) {
#if __has_builtin(__builtin_amdgcn_global_load_async_to_lds_b32)
  __builtin_amdgcn_global_load_async_to_lds_b32((AS1 int*)g, (AS3 int*)l, 0, 0);
#else
  unsigned loff = (unsigned)(unsigned long long)(AS3 char*)l;
  asm volatile("global_load_async_to_lds_b32 %0, %1, off" :: "v"(loff), "v"(g) : "memory");
#endif
}

__device__ __forceinline__ void wait_async0() {
#if __has_builtin(__builtin_amdgcn_s_wait_asynccnt)
  __builtin_amdgcn_s_wait_asynccnt(0);
#else
  asm volatile("s_wait_asynccnt 0" ::: "memory");
#endif
}

__global__ __launch_bounds__(NTHREADS) void ltc_fused_kernel(
    const float* __restrict__ inputs, const float* __restrict__ state,
    const float* __restrict__ input_w, const float* __restrict__ input_b,
    const float* __restrict__ sW, const float* __restrict__ sMu,
    const float* __restrict__ sSig, const float* __restrict__ sErev,
    const float* __restrict__ Wm, const float* __restrict__ Mu,
    const float* __restrict__ Sig, const float* __restrict__ Erev,
    const float* __restrict__ vleak, const float* __restrict__ gleak,
    const float* __restrict__ cm_t, float* __restrict__ out)
{
  // packed param tiles: {a=0.5*sigma, c=-0.5*sigma*mu, w=W, we=W*erev}
  __shared__ f4 rQ[TJ][H_SZ];      // 32 KB; raw staging union = float[8192]
  __shared__ f4 sQ[I_SZ][TJ];      //  8 KB; raw staging union = float[2048]
  __shared__ float xT[TB][I_SZ + 1];
  __shared__ float CDN[16];        // [0..7] = sum W rows, [8..15] = sum W*erev rows

  const int tid = threadIdx.x;
  const int bl  = tid & (TB - 1);  // lane -> b (wave32: one j per wave)
  const int jl  = tid >> 5;        // wave -> j
  const int b0  = blockIdx.x * TB;
  const int j0  = blockIdx.y * TJ;

  float* rraw = (float*)&rQ[0][0];
  float* sraw = (float*)&sQ[0][0];

  // ---- Phase 0: issue async-to-LDS copies of parameter tiles (raw) ----
  // recurrent rows j0..j0+7 of sigma, mu, W, erev are contiguous 8KB blocks
  {
    const char* g0 = (const char*)(Sig  + (size_t)j0 * H_SZ);
    const char* g1 = (const char*)(Mu   + (size_t)j0 * H_SZ);
    const char* g2 = (const char*)(Wm   + (size_t)j0 * H_SZ);
    const char* g3 = (const char*)(Erev + (size_t)j0 * H_SZ);
    char* l = (char*)rraw;
#pragma unroll
    for (int r = 0; r < 2; ++r) {
      int c = (tid + r * NTHREADS) * 16;          // 512 chunks of 16B per array
      async_b128(g0 + c, l + 0 * 8192 + c);
      async_b128(g1 + c, l + 1 * 8192 + c);
      async_b128(g2 + c, l + 2 * 8192 + c);
      async_b128(g3 + c, l + 3 * 8192 + c);
    }
  }
  // sensory columns j0..j0+7 (strided by H) -> [i][jj], b32 gathers
  {
#pragma unroll
    for (int r = 0; r < 2; ++r) {
      int e = tid + r * NTHREADS;                 // 512 elements per array
      int i = e >> 3, jj = e & 7;
      size_t go = (size_t)i * H_SZ + j0 + jj;
      async_b32(sSig  + go, sraw +    0 + e);
      async_b32(sMu   + go, sraw +  512 + e);
      async_b32(sW    + go, sraw + 1024 + e);
      async_b32(sErev + go, sraw + 1536 + e);
    }
  }

  // ---- x tile (computed: inputs*input_w+input_b), overlaps async copies ----
#pragma unroll
  for (int r = 0; r < 8; ++r) {
    int e = tid + NTHREADS * r;                   // 2048 elements, coalesced
    int bb = e >> 6, ii = e & 63;
    float x = inputs[(size_t)(b0 + bb) * I_SZ + ii];
    xT[bb][ii] = __builtin_fmaf(x, input_w[ii], input_b[ii]);
  }

  wait_async0();
  __syncthreads();

  // ---- transform raw -> packed f4 in place (register staging) ----
  float rs[8], rm[8], rw[8], re[8];
#pragma unroll
  for (int r = 0; r < 8; ++r) {
    int e = tid + NTHREADS * r;
    rs[r] = rraw[e];        rm[r] = rraw[2048 + e];
    rw[r] = rraw[4096 + e]; re[r] = rraw[6144 + e];
  }
  float qs[2], qm[2], qw[2], qe[2];
#pragma unroll
  for (int r = 0; r < 2; ++r) {
    int e = tid + NTHREADS * r;
    qs[r] = sraw[e];        qm[r] = sraw[512 + e];
    qw[r] = sraw[1024 + e]; qe[r] = sraw[1536 + e];
  }
  __syncthreads();
  f4* rQf = &rQ[0][0];
  f4* sQf = &sQ[0][0];
#pragma unroll
  for (int r = 0; r < 8; ++r) {
    int e = tid + NTHREADS * r;
    float a = 0.5f * rs[r];
    f4 q; q.x = a; q.y = -a * rm[r]; q.z = rw[r]; q.w = rw[r] * re[r];
    rQf[e] = q;
  }
#pragma unroll
  for (int r = 0; r < 2; ++r) {
    int e = tid + NTHREADS * r;
    float a = 0.5f * qs[r];
    f4 q; q.x = a; q.y = -a * qm[r]; q.z = qw[r]; q.w = qw[r] * qe[r];
    sQf[e] = q;
  }
  __syncthreads();

  // ---- per-j constants via WMMA f32 16x16x4: rows 0..7 <- W, rows 8..15 <- W*erev,
  //      B = all-ones (layout independent) => D[m,0] = row sums. Wave 0 only (EXEC all-1).
#if HAVE_WMMA4
  if (tid < 32) {
    int m  = tid & 15;
    int hi = tid >> 4;                    // lanes 16..31 hold K=2,3
    int jm = (m < 8) ? m : (m - 8);
    bool useW = (m < 8);
    f2 bones; bones.x = 1.0f; bones.y = 1.0f;
    f8 d = {0.f, 0.f, 0.f, 0.f, 0.f, 0.f, 0.f, 0.f};
    for (int k0 = 0; k0 < H_SZ; k0 += 4) {
      int kk = k0 + hi * 2;
      f4 q0 = rQ[jm][kk], q1 = rQ[jm][kk + 1];
      f2 av; av.x = useW ? q0.z : q0.w; av.y = useW ? q1.z : q1.w;
      d = __builtin_amdgcn_wmma_f32_16x16x4_f32(false, av, false, bones,
                                                (short)0, d, false, false);
    }
    for (int i0 = 0; i0 < I_SZ; i0 += 4) {
      int kk = i0 + hi * 2;
      f4 q0 = sQ[kk][jm], q1 = sQ[kk + 1][jm];
      f2 av; av.x = useW ? q0.z : q0.w; av.y = useW ? q1.z : q1.w;
      d = __builtin_amdgcn_wmma_f32_16x16x4_f32(false, av, false, bones,
                                                (short)0, d, false, false);
    }
    if (m == 0) {                          // lane0: rows 0..7 (CD); lane16: rows 8..15 (CN)
      float* dst = &CDN[hi * 8];
#pragma unroll
      for (int r = 0; r < 8; ++r) dst[r] = d[r];
    }
  }
#else
  if (tid < TJ) {
    float cd = 0.f, cn = 0.f;
    for (int k = 0; k < H_SZ; ++k) { f4 q = rQ[tid][k]; cd += q.z; cn += q.w; }
    for (int i = 0; i < I_SZ; ++i) { f4 q = sQ[i][tid]; cd += q.z; cn += q.w; }
    CDN[tid] = cd; CDN[8 + tid] = cn;
  }
#endif
  __syncthreads();

  // ---- sensory accumulation (per b,j; computed once) ----
  const int j = j0 + jl;
  const int b = b0 + bl;
  float accSN = 0.f, accSD = 0.f;
#pragma unroll 4
  for (int i = 0; i < I_SZ; ++i) {
    f4 q = sQ[i][jl];                      // broadcast within wave
    float xv = xT[bl][i];                  // pad 65 -> conflict-free
    float t = fast_tanh(__builtin_fmaf(xv, q.x, q.y));
    accSN = __builtin_fmaf(q.w, t, accSN);
    accSD = __builtin_fmaf(q.z, t, accSD);
  }

  float cm = cm_t[j], gl = gleak[j], vl = vleak[j];
  // sigmoid = 0.5*(1+tanh): fold all constant halves into bases
  float baseD = cm + gl + 0.5f * (accSD + CDN[jl]);
  float baseN = gl * vl + 0.5f * (accSN + CDN[8 + jl]);
  float v = state[(size_t)b * H_SZ + j];

  // ---- 6 fused ODE unfolds: each (b,j) independent ----
  for (int u = 0; u < UNFOLDS; ++u) {
    float aN = 0.f, aD = 0.f;
#pragma unroll 8
    for (int k = 0; k < H_SZ; ++k) {
      f4 q = rQ[jl][k];                    // one broadcast ds_load_b128 per term
      float t = fast_tanh(__builtin_fmaf(v, q.x, q.y));
      aN = __builtin_fmaf(q.w, t, aN);
      aD = __builtin_fmaf(q.z, t, aD);
    }
    float num = __builtin_fmaf(0.5f, aN, __builtin_fmaf(cm, v, baseN));
    float den = __builtin_fmaf(0.5f, aD, baseD);
    v = num / den;
  }

  out[(size_t)b * H_SZ + j] = v;
}

extern "C" void kernel_launch(void* const* d_in, const int* in_sizes, int n_in,
                              void* d_out, int out_size, void* d_ws, size_t ws_size,
                              hipStream_t stream) {
  (void)in_sizes; (void)n_in; (void)out_size; (void)d_ws; (void)ws_size;
  const float* inputs  = (const float*)d_in[0];
  const float* state   = (const float*)d_in[1];
  const float* input_w = (const float*)d_in[2];
  const float* input_b = (const float*)d_in[3];
  const float* sW      = (const float*)d_in[4];
  const float* sMu     = (const float*)d_in[5];
  const float* sSig    = (const float*)d_in[6];
  const float* sErev   = (const float*)d_in[7];
  const float* Wm      = (const float*)d_in[8];
  const float* Mu      = (const float*)d_in[9];
  const float* Sig     = (const float*)d_in[10];
  const float* Erev    = (const float*)d_in[11];
  const float* vleak   = (const float*)d_in[12];
  const float* gleak   = (const float*)d_in[13];
  const float* cm_t    = (const float*)d_in[14];
  float* out = (float*)d_out;

  dim3 grid(B_SZ / TB, H_SZ / TJ);   // (32, 32)
  dim3 block(NTHREADS);
  hipLaunchKernelGGL(ltc_fused_kernel, grid, block, 0, stream,
                     inputs, state, input_w, input_b, sW, sMu, sSig, sErev,
                     Wm, Mu, Sig, Erev, vleak, gleak, cm_t, out);
}